// Non_local_12197707120960
// MI455X (gfx1250) — compile-verified
//
#include <hip/hip_runtime.h>
#include <hip/hip_bf16.h>

typedef __attribute__((ext_vector_type(2))) float v2f;
typedef __attribute__((ext_vector_type(8))) float v8f;

#define BATCH 32
#define CCH   512
#define NPIX  1152   // 48*24

// ---------------------------------------------------------------------------
// Kernel 1: projection GEMM  [16 x C] x [C x N] -> g/theta/phi rows via
// V_WMMA_F32_16X16X4_F32 (exact fp32). One wave per 16-column tile.
// ---------------------------------------------------------------------------
__global__ __launch_bounds__(256)
void nl_proj_wmma(const float* __restrict__ x,
                  const float* __restrict__ g_w,  const float* __restrict__ g_b,
                  const float* __restrict__ th_w, const float* __restrict__ th_b,
                  const float* __restrict__ ph_w, const float* __restrict__ ph_b,
                  float* __restrict__ gx, float* __restrict__ tx,
                  float* __restrict__ px)
{
    // Weight matrix staged in LDS: Wlds[(k>>1)*32 + m*2 + (k&1)] = Wmat[m][k]
    // so each lane reads its (K, K+1) A-pair as one contiguous b64.
    __shared__ float Wlds[16 * CCH];
    for (int i = threadIdx.x; i < 16 * CCH; i += blockDim.x) {
        int m = i >> 9;          // row 0..15
        int k = i & (CCH - 1);   // 0..511
        float v = 0.0f;
        if (m == 0)      v = g_w[k];
        else if (m == 1) v = th_w[k];
        else if (m == 2) v = ph_w[k];
        Wlds[(k >> 1) * 32 + (m << 1) + (k & 1)] = v;
    }
    __syncthreads();

    const int lane = threadIdx.x & 31;
    const int wave = blockIdx.x * (blockDim.x >> 5) + (threadIdx.x >> 5);
    const int tiles_per_b = NPIX / 16;            // 72
    if (wave >= BATCH * tiles_per_b) return;      // wave-uniform branch

    const int b    = wave / tiles_per_b;
    const int n0   = (wave % tiles_per_b) * 16;
    const int m    = lane & 15;                   // A row / B column-in-tile
    const int half = lane >> 4;                   // K sub-pair selector

    const float* xcol = x + (size_t)b * CCH * NPIX + n0 + m;

    v8f acc0 = {0.f,0.f,0.f,0.f,0.f,0.f,0.f,0.f};
    v8f acc1 = {0.f,0.f,0.f,0.f,0.f,0.f,0.f,0.f};

    #pragma unroll 4
    for (int k0 = 0; k0 < CCH; k0 += 8) {
        // ---- step A: K = k0..k0+3 ----
        {
            const int ka = k0 + 2 * half;
            v2f A, Bm;
            A.x = Wlds[(ka >> 1) * 32 + (m << 1) + 0];
            A.y = Wlds[(ka >> 1) * 32 + (m << 1) + 1];
            const float* p = xcol + (size_t)ka * NPIX;
            Bm.x = p[0];
            Bm.y = p[NPIX];
            acc0 = __builtin_amdgcn_wmma_f32_16x16x4_f32(
                false, A, false, Bm, (short)0, acc0, false, false);
        }
        // ---- step B: K = k0+4..k0+7 (second accumulator for ILP) ----
        {
            const int ka = k0 + 4 + 2 * half;
            v2f A, Bm;
            A.x = Wlds[(ka >> 1) * 32 + (m << 1) + 0];
            A.y = Wlds[(ka >> 1) * 32 + (m << 1) + 1];
            const float* p = xcol + (size_t)ka * NPIX;
            Bm.x = p[0];
            Bm.y = p[NPIX];
            acc1 = __builtin_amdgcn_wmma_f32_16x16x4_f32(
                false, A, false, Bm, (short)0, acc1, false, false);
        }
    }

    // D layout: lanes 0-15 hold N=n0+lane, VGPR r = row M=r.
    if (lane < 16) {
        const size_t off = (size_t)b * NPIX + n0 + lane;
        gx[off] = acc0[0] + acc1[0] + g_b[0];
        tx[off] = acc0[1] + acc1[1] + th_b[0];
        px[off] = acc0[2] + acc1[2] + ph_b[0];
    }
}

// ---------------------------------------------------------------------------
// Kernel 2: s[b] = sum_n phi[b,n] * g[b,n]   (one block per batch)
// ---------------------------------------------------------------------------
__global__ __launch_bounds__(256)
void nl_reduce_s(const float* __restrict__ gx, const float* __restrict__ px,
                 float* __restrict__ s)
{
    const int b = blockIdx.x;
    const float* gp = gx + (size_t)b * NPIX;
    const float* pp = px + (size_t)b * NPIX;

    float acc = 0.0f;
    for (int n = threadIdx.x; n < NPIX; n += blockDim.x)
        acc += gp[n] * pp[n];

    // wave32 reduction
    #pragma unroll
    for (int off = 16; off > 0; off >>= 1)
        acc += __shfl_down(acc, off, 32);

    __shared__ float red[8];
    const int widx = threadIdx.x >> 5;
    const int lane = threadIdx.x & 31;
    if (lane == 0) red[widx] = acc;
    __syncthreads();
    if (threadIdx.x == 0) {
        float t = 0.0f;
        #pragma unroll
        for (int w = 0; w < 8; ++w) t += red[w];
        s[b] = t;
    }
}

// ---------------------------------------------------------------------------
// Kernel 3: fused epilogue  z = x + alpha_c * (theta[b,n]*s_b/N) + kappa_c
// float4-vectorized streaming kernel (bandwidth bound part).
// ---------------------------------------------------------------------------
__global__ __launch_bounds__(256)
void nl_epilogue(const float* __restrict__ x,
                 const float* __restrict__ tx, const float* __restrict__ s,
                 const float* __restrict__ W_w, const float* __restrict__ W_b,
                 const float* __restrict__ bn_g, const float* __restrict__ bn_b,
                 const float* __restrict__ bn_m, const float* __restrict__ bn_v,
                 float* __restrict__ z)
{
    const size_t total4 = (size_t)BATCH * CCH * NPIX / 4;
    const size_t i4 = (size_t)blockIdx.x * blockDim.x + threadIdx.x;
    if (i4 >= total4) return;
    const size_t i = i4 * 4;

    const int n = (int)(i % NPIX);                 // NPIX % 4 == 0 -> 4 elems share b,c
    const int c = (int)((i / NPIX) % CCH);
    const int b = (int)(i / ((size_t)CCH * NPIX));

    const float scale = __frsqrt_rn(bn_v[c] + 1e-5f) * bn_g[c];
    const float alpha = scale * W_w[c];
    const float kappa = scale * (W_b[c] - bn_m[c]) + bn_b[c];
    const float sb    = s[b] * (1.0f / (float)NPIX);

    const float4 xv = *(const float4*)(x + i);
    const float4 tv = *(const float4*)(tx + (size_t)b * NPIX + n);

    float4 o;
    o.x = fmaf(alpha, tv.x * sb, xv.x + kappa);
    o.y = fmaf(alpha, tv.y * sb, xv.y + kappa);
    o.z = fmaf(alpha, tv.z * sb, xv.z + kappa);
    o.w = fmaf(alpha, tv.w * sb, xv.w + kappa);
    *(float4*)(z + i) = o;
}

// ---------------------------------------------------------------------------
extern "C" void kernel_launch(void* const* d_in, const int* in_sizes, int n_in,
                              void* d_out, int out_size, void* d_ws, size_t ws_size,
                              hipStream_t stream)
{
    const float* x    = (const float*)d_in[0];
    const float* g_w  = (const float*)d_in[1];
    const float* g_b  = (const float*)d_in[2];
    const float* th_w = (const float*)d_in[3];
    const float* th_b = (const float*)d_in[4];
    const float* ph_w = (const float*)d_in[5];
    const float* ph_b = (const float*)d_in[6];
    const float* W_w  = (const float*)d_in[7];
    const float* W_b  = (const float*)d_in[8];
    const float* bn_g = (const float*)d_in[9];
    const float* bn_b = (const float*)d_in[10];
    const float* bn_m = (const float*)d_in[11];
    const float* bn_v = (const float*)d_in[12];
    float* z = (float*)d_out;

    float* gx = (float*)d_ws;                    // B*N
    float* tx = gx + (size_t)BATCH * NPIX;       // B*N
    float* px = tx + (size_t)BATCH * NPIX;       // B*N
    float* s  = px + (size_t)BATCH * NPIX;       // B

    // K1: 32 batches * 72 tiles = 2304 waves; 8 waves/block -> 288 blocks
    nl_proj_wmma<<<288, 256, 0, stream>>>(x, g_w, g_b, th_w, th_b, ph_w, ph_b,
                                          gx, tx, px);

    // K2: one block per batch
    nl_reduce_s<<<BATCH, 256, 0, stream>>>(gx, px, s);

    // K3: streaming epilogue over B*C*N/4 float4s
    const size_t total4 = (size_t)BATCH * CCH * NPIX / 4;
    const int blocks = (int)((total4 + 255) / 256);
    nl_epilogue<<<blocks, 256, 0, stream>>>(x, tx, s, W_w, W_b,
                                            bn_g, bn_b, bn_m, bn_v, z);
}